// MultiHeadGcnGVAE_3513283248869
// MI455X (gfx1250) — compile-verified
//
#include <hip/hip_runtime.h>
#include <hip/hip_bf16.h>
#include <math.h>
#include <stdint.h>

typedef __attribute__((ext_vector_type(2))) float v2f;
typedef __attribute__((ext_vector_type(8))) float v8f;

#define NUM_HEADS 3
#define F_IN 512
#define H1 256
#define H2 64

// Toggle: CDNA5 async global->LDS staging (ASYNCcnt path). Set to 0 to fall
// back to plain LDS stores if the assembler rejects the mnemonic.
#define USE_ASYNC_LDS 1

// ---------------- degree / normalization ----------------

__global__ void k_deg_init(float* deg, int n) {
    int i = blockIdx.x * blockDim.x + threadIdx.x;
    if (i < n) deg[i] = 1.0f;   // self loop
}

__global__ void k_deg_acc(const int* __restrict__ dst, float* deg, int E) {
    int e = blockIdx.x * blockDim.x + threadIdx.x;
    if (e < E) atomicAdd(&deg[dst[e]], 1.0f);
}

__global__ void k_invsqrt(float* deg, int n) {
    int i = blockIdx.x * blockDim.x + threadIdx.x;
    if (i < n) deg[i] = rsqrtf(deg[i]);
}

__global__ void k_edge_norm(const int* __restrict__ src, const int* __restrict__ dst,
                            const float* __restrict__ invs, float* __restrict__ enorm, int E) {
    int e = blockIdx.x * blockDim.x + threadIdx.x;
    if (e < E) enorm[e] = invs[src[e]] * invs[dst[e]];
}

// ---------------- generic batched WMMA GEMM (fp32, 16x16x4) ----------------
// C[head] = A[head] (MxK, row major) @ B[head] (KxN, row major)
// One wave32 per 16x16 output tile. EXEC is all ones (no divergence).

__global__ __launch_bounds__(32)
void k_gemm_wmma(const float* __restrict__ A, const float* __restrict__ B,
                 float* __restrict__ C, int M, int N, int K,
                 long long sA, long long sB, long long sC) {
    const float* Ah = A + (long long)blockIdx.z * sA;
    const float* Bh = B + (long long)blockIdx.z * sB;
    float*       Ch = C + (long long)blockIdx.z * sC;

    const int lane = threadIdx.x & 31;
    const int m16  = lane & 15;      // M (for A) / N (for B,C,D)
    const int half = lane >> 4;      // selects K pair / M upper half

    const int rowBase = blockIdx.y << 4;
    const int colBase = blockIdx.x << 4;

    const float* arow = Ah + (long long)(rowBase + m16) * K;

    v8f c = {};
    for (int k0 = 0; k0 < K; k0 += 4) {
        const int ka = k0 + 2 * half;
        __builtin_prefetch(arow + ka + 32, 0, 1);   // global_prefetch_b8
        v2f a;
        a.x = arow[ka];
        a.y = arow[ka + 1];
        v2f b;
        b.x = Bh[(long long)ka       * N + colBase + m16];
        b.y = Bh[(long long)(ka + 1) * N + colBase + m16];
        c = __builtin_amdgcn_wmma_f32_16x16x4_f32(false, a, false, b,
                                                  (short)0, c, false, false);
    }
#pragma unroll
    for (int r = 0; r < 8; ++r) {
        const int row = rowBase + r + 8 * half;
        Ch[(long long)row * N + colBase + m16] = c[r];
    }
}

// ---------------- propagation: agg = h*self_w + bias, then scatter edges ----------------

__global__ void k_init_agg(const float* __restrict__ Hpre, const float* __restrict__ invs,
                           const float* __restrict__ bias, float* __restrict__ Agg,
                           int nN, int F) {
    long long i = (long long)blockIdx.x * blockDim.x + threadIdx.x;   // over N*F
    const long long total = (long long)nN * F;
    if (i >= total) return;
    const int n = (int)(i / F);
    const int f = (int)(i % F);
    const int h = blockIdx.y;
    const float sw = invs[n] * invs[n];      // self-loop weight 1/deg
    const long long off = (long long)h * total + i;
    Agg[off] = Hpre[off] * sw + bias[h * F + f];
}

__global__ void k_scatter(const int* __restrict__ src, const int* __restrict__ dst,
                          const float* __restrict__ enorm,
                          const float* __restrict__ Hpre, float* __restrict__ Agg,
                          int E, int F, long long headStride) {
    long long i = (long long)blockIdx.x * blockDim.x + threadIdx.x;
    const int nChunk = F >> 2;                        // float4 chunks
    const long long total = (long long)E * nChunk;
    if (i >= total) return;
    const int e = (int)(i / nChunk);
    const int f = (int)((i % nChunk) << 2);
    const int h = blockIdx.y;
    const float w = enorm[e];
    const float4 v = *(const float4*)(Hpre + (long long)h * headStride +
                                      (long long)src[e] * F + f);
    float* ag = Agg + (long long)h * headStride + (long long)dst[e] * F + f;
    atomicAdd(ag + 0, v.x * w);
    atomicAdd(ag + 1, v.y * w);
    atomicAdd(ag + 2, v.z * w);
    atomicAdd(ag + 3, v.w * w);
}

__global__ void k_relu(float* a, long long total) {
    long long i = (long long)blockIdx.x * blockDim.x + threadIdx.x;
    if (i < total) a[i] = fmaxf(a[i], 0.0f);
}

// ---------------- reparametrize + head-max ----------------

__device__ __forceinline__ unsigned pcg_hash(unsigned v) {
    v = v * 747796405u + 2891336453u;
    unsigned w = ((v >> ((v >> 28) + 4u)) ^ v) * 277803737u;
    return (w >> 22) ^ w;
}

__device__ __forceinline__ float gauss_hash(unsigned seed) {
    const unsigned h1 = pcg_hash(seed);
    const unsigned h2 = pcg_hash(seed ^ 0x9e3779b9u);
    const float u1 = (float)(h1 + 1u) * 2.3283064e-10f;   // (0,1]
    const float u2 = (float)h2 * 2.3283064e-10f;          // [0,1)
    const float r = sqrtf(-2.0f * __logf(u1));
    return r * __cosf(6.283185307f * u2);
}

__global__ void k_z_max(const float* __restrict__ mu, const float* __restrict__ lv,
                        float* __restrict__ z_max,
                        float* __restrict__ mu_max, float* __restrict__ lv_max,
                        int nN, int F) {
    long long i = (long long)blockIdx.x * blockDim.x + threadIdx.x;   // over N*F
    const long long hs = (long long)nN * F;
    if (i >= hs) return;
    float mm = -INFINITY, lm = -INFINITY, zm = -INFINITY;
#pragma unroll
    for (int h = 0; h < NUM_HEADS; ++h) {
        const float m = mu[h * hs + i];
        const float l = lv[h * hs + i];
        const float eps = gauss_hash((unsigned)(i * NUM_HEADS + h) * 2654435761u + 12345u);
        const float z = m + eps * __expf(l);
        mm = fmaxf(mm, m);
        lm = fmaxf(lm, l);
        zm = fmaxf(zm, z);
    }
    z_max[i] = zm;
    mu_max[i] = mm;
    lv_max[i] = lm;
}

// ---------------- adjacency reconstruction: sigmoid(Z @ Z^T) ----------------
// Block = 256 threads (8 wave32) computes a 128x128 output tile.
// A/B panels (128x64 f32 each) staged into LDS via CDNA5 async global->LDS
// copies (ASYNCcnt), padded to 68 floats/row: rows stay 16B aligned for b128
// and lanes land in distinct banks for the WMMA fragment reads.
// Each wave owns a 16x128 strip (8 accumulators) -> 8x A-fragment reuse.

#define RT 128                 // tile edge
#define LDP 68                 // padded LDS row (floats)

__device__ __forceinline__ void stage_panel_async(const float* __restrict__ gsrc,
                                                  float* __restrict__ lrow,
                                                  int Kd) {
    // copy one float4: gsrc -> lrow (LDS), via async DMA or plain store
#if USE_ASYNC_LDS
    unsigned lds_off = (unsigned)(uintptr_t)lrow;      // ISA: generic LDS addr low 32b = LDS offset
    unsigned long long ga = (unsigned long long)(uintptr_t)gsrc;
    asm volatile("global_load_async_to_lds_b128 %0, %1, off"
                 :: "v"(lds_off), "v"(ga) : "memory");
#else
    *(float4*)lrow = *(const float4*)gsrc;
#endif
    (void)Kd;
}

__global__ __launch_bounds__(256)
void k_recon_wmma(const float* __restrict__ Z, float* __restrict__ out, int nN, int Kd) {
    __shared__ float As[RT][LDP];
    __shared__ float Bs[RT][LDP];

    const int tid  = threadIdx.x;
    const int w    = tid >> 5;          // wave id 0..7
    const int lane = tid & 31;
    const int m16  = lane & 15;
    const int half = lane >> 4;

    const int rowBase = blockIdx.y * RT;
    const int colBase = blockIdx.x * RT;

    // ---- stage A (rows) and B (cols) panels: 128x64 floats each ----
    // 2048 float4 chunks per panel, 256 threads -> 8 chunks each
#pragma unroll
    for (int i = 0; i < 8; ++i) {
        const int c    = i * 256 + tid;
        const int row  = c >> 4;            // 16 float4 per 64-float row
        const int col4 = (c & 15) << 2;
        stage_panel_async(Z + (long long)(rowBase + row) * Kd + col4, &As[row][col4], Kd);
        stage_panel_async(Z + (long long)(colBase + row) * Kd + col4, &Bs[row][col4], Kd);
    }
#if USE_ASYNC_LDS
    asm volatile("s_wait_asynccnt 0" ::: "memory");
#endif
    __syncthreads();

    // ---- compute: wave w -> rows [w*16, w*16+16), all 8 column tiles ----
    v8f acc[8] = {};
    for (int k0 = 0; k0 < H2; k0 += 4) {
        const int ka = k0 + 2 * half;
        v2f a;
        a.x = As[w * 16 + m16][ka];
        a.y = As[w * 16 + m16][ka + 1];
#pragma unroll
        for (int t = 0; t < 8; ++t) {
            v2f b;
            b.x = Bs[t * 16 + m16][ka];
            b.y = Bs[t * 16 + m16][ka + 1];
            acc[t] = __builtin_amdgcn_wmma_f32_16x16x4_f32(false, a, false, b,
                                                           (short)0, acc[t], false, false);
        }
    }

    // ---- epilogue: sigmoid + non-temporal streaming stores (268 MB total) ----
#pragma unroll
    for (int t = 0; t < 8; ++t) {
#pragma unroll
        for (int r = 0; r < 8; ++r) {
            const int row = rowBase + w * 16 + r + 8 * half;
            const int col = colBase + t * 16 + m16;
            const float s = 1.0f / (1.0f + __expf(-acc[t][r]));
            __builtin_nontemporal_store(s, &out[(long long)row * nN + col]);
        }
    }
}

// ---------------- host driver ----------------

extern "C" void kernel_launch(void* const* d_in, const int* in_sizes, int n_in,
                              void* d_out, int out_size, void* d_ws, size_t ws_size,
                              hipStream_t stream) {
    const float* x   = (const float*)d_in[0];
    const int*   ei  = (const int*)  d_in[1];
    const float* W0  = (const float*)d_in[2];
    const float* b0  = (const float*)d_in[3];
    const float* Wmu = (const float*)d_in[4];
    const float* bmu = (const float*)d_in[5];
    const float* Wlv = (const float*)d_in[6];
    const float* blv = (const float*)d_in[7];

    const int nN = in_sizes[0] / F_IN;   // 8192
    const int E  = in_sizes[1] / 2;      // 262144
    const int* srcI = ei;
    const int* dstI = ei + E;

    // workspace carve-up (floats)
    float* ws = (float*)d_ws;
    float* invs  = ws; ws += nN;                                  // deg -> inv_sqrt (in place)
    float* enorm = ws; ws += E;
    float* hpre  = ws; ws += (long long)NUM_HEADS * nN * H1;
    float* hagg  = ws; ws += (long long)NUM_HEADS * nN * H1;      // becomes h (post relu)
    float* mupre = ws; ws += (long long)NUM_HEADS * nN * H2;
    float* lvpre = ws; ws += (long long)NUM_HEADS * nN * H2;
    float* muagg = ws; ws += (long long)NUM_HEADS * nN * H2;
    float* lvagg = ws; ws += (long long)NUM_HEADS * nN * H2;
    float* zmax  = ws; ws += (long long)nN * H2;

    float* adj    = (float*)d_out;
    float* mu_max = adj + (long long)nN * nN;
    float* lv_max = mu_max + (long long)nN * H2;

    const long long sH1 = (long long)nN * H1;
    const long long sH2 = (long long)nN * H2;

    // 1) GCN normalization
    k_deg_init <<<(nN + 255) / 256, 256, 0, stream>>>(invs, nN);
    k_deg_acc  <<<(E + 255) / 256, 256, 0, stream>>>(dstI, invs, E);
    k_invsqrt  <<<(nN + 255) / 256, 256, 0, stream>>>(invs, nN);
    k_edge_norm<<<(E + 255) / 256, 256, 0, stream>>>(srcI, dstI, invs, enorm, E);

    // 2) layer 1: h_pre = x @ W0   (WMMA f32)
    k_gemm_wmma<<<dim3(H1 / 16, nN / 16, NUM_HEADS), 32, 0, stream>>>(
        x, W0, hpre, nN, H1, F_IN, 0LL, (long long)F_IN * H1, sH1);
    k_init_agg<<<dim3((unsigned)((sH1 + 255) / 256), NUM_HEADS), 256, 0, stream>>>(
        hpre, invs, b0, hagg, nN, H1);
    k_scatter <<<dim3((unsigned)(((long long)E * (H1 / 4) + 255) / 256), NUM_HEADS), 256, 0, stream>>>(
        srcI, dstI, enorm, hpre, hagg, E, H1, sH1);
    k_relu    <<<(unsigned)((NUM_HEADS * sH1 + 255) / 256), 256, 0, stream>>>(
        hagg, NUM_HEADS * sH1);

    // 3) mu head
    k_gemm_wmma<<<dim3(H2 / 16, nN / 16, NUM_HEADS), 32, 0, stream>>>(
        hagg, Wmu, mupre, nN, H2, H1, sH1, (long long)H1 * H2, sH2);
    k_init_agg<<<dim3((unsigned)((sH2 + 255) / 256), NUM_HEADS), 256, 0, stream>>>(
        mupre, invs, bmu, muagg, nN, H2);
    k_scatter <<<dim3((unsigned)(((long long)E * (H2 / 4) + 255) / 256), NUM_HEADS), 256, 0, stream>>>(
        srcI, dstI, enorm, mupre, muagg, E, H2, sH2);

    // 4) logvar head
    k_gemm_wmma<<<dim3(H2 / 16, nN / 16, NUM_HEADS), 32, 0, stream>>>(
        hagg, Wlv, lvpre, nN, H2, H1, sH1, (long long)H1 * H2, sH2);
    k_init_agg<<<dim3((unsigned)((sH2 + 255) / 256), NUM_HEADS), 256, 0, stream>>>(
        lvpre, invs, blv, lvagg, nN, H2);
    k_scatter <<<dim3((unsigned)(((long long)E * (H2 / 4) + 255) / 256), NUM_HEADS), 256, 0, stream>>>(
        srcI, dstI, enorm, lvpre, lvagg, E, H2, sH2);

    // 5) reparametrize + head max (writes mu_max / logvar_max directly to d_out)
    k_z_max<<<(unsigned)((sH2 + 255) / 256), 256, 0, stream>>>(
        muagg, lvagg, zmax, mu_max, lv_max, nN, H2);

    // 6) adjacency reconstruction (dominant bytes: 268 MB NT stores)
    k_recon_wmma<<<dim3(nN / RT, nN / RT), 256, 0, stream>>>(zmax, adj, nN, H2);
}